// VectorQuantizer_16406775070747
// MI455X (gfx1250) — compile-verified
//
#include <hip/hip_runtime.h>

// ---------------------------------------------------------------------------
// Vector-quantizer (VQ-VAE codebook lookup) for gfx1250 / MI455X.
//   inputs  : (16, 64, 64, 64) f32, BCHW  -> N=65536 rows of C=64
//   codebook: (1024, 64) f32
//   out     : zis (65536) as float, then zqs (16,64,64,64) f32 BCHW
// Matmul on V_WMMA_F32_16X16X4_F32; distances + argmin on VALU; codebook
// staged via LDS (pitch-65 padding for conflict-free B-fragment loads).
// ---------------------------------------------------------------------------

typedef __attribute__((ext_vector_type(2))) float v2f;
typedef __attribute__((ext_vector_type(8))) float v8f;

#define NUM_EMB     1024
#define EDIM        64
#define HW          4096      // 64*64
#define CHW         262144    // 64*HW
#define N_TOTAL     65536     // 16*HW
#define STAGE_CODES 128       // codes per LDS stage
#define N_STAGES    (NUM_EMB / STAGE_CODES)
#define CB_PITCH    65        // 64 + 1 pad -> no LDS bank conflicts

__global__ __launch_bounds__(256)
void vq_wmma_kernel(const float* __restrict__ in,
                    const float* __restrict__ cbg,
                    float* __restrict__ zis,
                    float* __restrict__ zqs)
{
    __shared__ float cb[STAGE_CODES * CB_PITCH];   // staged codebook tile
    __shared__ float cbNorm[STAGE_CODES];          // ||e_k||^2 for staged codes
    __shared__ float rowNormS[8][16];              // per-wave ||z_n||^2
    __shared__ int   widx[8][16];                  // per-wave final indices

    const int t    = threadIdx.x;       // 0..255  (8 waves of 32)
    const int wave = t >> 5;
    const int lane = t & 31;
    const int n16  = lane & 15;         // M-row (A) / N-col (B,D) within tile
    const int hi   = lane >> 4;         // lane half

    // this wave's 16 consecutive input rows (never cross a batch boundary:
    // 4096 % 16 == 0)
    const int n0   = blockIdx.x * 128 + wave * 16;
    const int bimg = n0 >> 12;
    const int hw0  = n0 & 4095;

    // ---- load A fragments (16 rows x 64 ch) in WMMA 16x16x4 f32 layout ----
    // a[2j]   : lanes 0-15 -> (M=lane, K=4j+0) ; lanes 16-31 -> (M, K=4j+2)
    // a[2j+1] : K = 4j+1 / 4j+3
    // memory is channel-major (BCHW): fixed c => 16 consecutive rows are
    // contiguous (64B per half-wave) -> coalesced.
    const float* abase = in + (size_t)bimg * CHW + hw0 + n16;
    float a[32];
#pragma unroll
    for (int j = 0; j < 16; ++j) {
        const int c0 = 4 * j + 2 * hi;
        a[2 * j]     = abase[(size_t)(c0)     * HW];
        a[2 * j + 1] = abase[(size_t)(c0 + 1) * HW];
    }

    // ---- row norms ||z||^2 : lane l holds half the channels of row l,
    // lane l+16 the other half -> one xor-16 shuffle completes the sum.
    float pn = 0.f;
#pragma unroll
    for (int i = 0; i < 32; ++i) pn += a[i] * a[i];
    pn += __shfl_xor(pn, 16, 32);
    if (lane < 16) rowNormS[wave][n16] = pn;

    // each lane needs norms of its 8 D-rows (M = r + 8*hi)
    float rn[8];
    {
        const int rb = hi * 8;
#pragma unroll
        for (int r = 0; r < 8; ++r) rn[r] = rowNormS[wave][rb + r];
    }

    float minv[8];
    int   mini[8];
#pragma unroll
    for (int r = 0; r < 8; ++r) { minv[r] = 3.402823e38f; mini[r] = 0; }

    // =============== stage loop over codebook (8 x 128 codes) ==============
    for (int s = 0; s < N_STAGES; ++s) {
        __syncthreads();
        const float* src = cbg + (size_t)s * STAGE_CODES * EDIM;
        // prefetch next stage while we still have this one to chew on
        if (s + 1 < N_STAGES)
            __builtin_prefetch(src + STAGE_CODES * EDIM + t * 32, 0, 1);
        // 8192 floats by 256 threads, padded pitch in LDS
#pragma unroll
        for (int i = 0; i < 32; ++i) {
            const int linear = i * 256 + t;
            const int code   = linear >> 6;
            const int c      = linear & 63;
            cb[code * CB_PITCH + c] = src[linear];
        }
        __syncthreads();
        if (t < STAGE_CODES) {
            float s2 = 0.f;
#pragma unroll 8
            for (int c = 0; c < 64; ++c) {
                const float v = cb[t * CB_PITCH + c];
                s2 += v * v;
            }
            cbNorm[t] = s2;
        }
        __syncthreads();

        // -------- 8 column tiles of 16 codes each --------
#pragma unroll 1
        for (int tile = 0; tile < 8; ++tile) {
            // B fragment source: row = code (N = n16), col = channel.
            // pitch 65 => lanes 0..15 hit 16 distinct banks.
            const float* brow = &cb[(tile * 16 + n16) * CB_PITCH + hi];
            const float  cn   = cbNorm[tile * 16 + n16];

            v8f acc = {};
#pragma unroll
            for (int j = 0; j < 16; ++j) {
                v2f av; av.x = a[2 * j]; av.y = a[2 * j + 1];
                v2f bv; bv.x = brow[4 * j]; bv.y = brow[4 * j + 2];
                acc = __builtin_amdgcn_wmma_f32_16x16x4_f32(
                        false, av, false, bv, (short)0, acc, false, false);
            }

            const int col = s * STAGE_CODES + tile * 16 + n16;
#pragma unroll
            for (int r = 0; r < 8; ++r) {
                const float dist = rn[r] + cn - 2.0f * acc[r];
                if (dist < minv[r]) { minv[r] = dist; mini[r] = col; }
            }
        }
    }

    // ---- cross-lane argmin over the 16-lane N dimension (per half) ----
#pragma unroll
    for (int r = 0; r < 8; ++r) {
        float v = minv[r];
        int   i = mini[r];
#pragma unroll
        for (int m = 8; m >= 1; m >>= 1) {
            const float ov = __shfl_xor(v, m, 32);
            const int   oi = __shfl_xor(i, m, 32);
            if (ov < v || (ov == v && oi < i)) { v = ov; i = oi; }
        }
        minv[r] = v;
        mini[r] = i;
    }

    // lane 0 holds rows 0..7, lane 16 holds rows 8..15
    if (n16 == 0) {
        const int rb = hi * 8;
#pragma unroll
        for (int r = 0; r < 8; ++r) {
            widx[wave][rb + r]  = mini[r];
            zis[n0 + rb + r]    = (float)mini[r];
        }
    }
    __syncthreads();

    // ---- zqs: gather selected codes, write BCHW-coalesced ----
    const int    myidx = widx[wave][n16];
    const float* crow  = cbg + (size_t)myidx * EDIM;
    float*       obase = zqs + (size_t)bimg * CHW + hw0 + n16;
#pragma unroll
    for (int it = 0; it < 32; ++it) {
        const int c = it * 2 + hi;           // half-wave per channel
        obase[(size_t)c * HW] = crow[c];     // 64B contiguous per 16 lanes
    }
}

extern "C" void kernel_launch(void* const* d_in, const int* in_sizes, int n_in,
                              void* d_out, int out_size, void* d_ws, size_t ws_size,
                              hipStream_t stream) {
    const float* in  = (const float*)d_in[0];   // (16,64,64,64) f32
    const float* cbg = (const float*)d_in[1];   // (1024,64) f32
    float* zis = (float*)d_out;                 // 65536 indices (as float)
    float* zqs = (float*)d_out + N_TOTAL;       // 4194304 f32
    vq_wmma_kernel<<<N_TOTAL / 128, 256, 0, stream>>>(in, cbg, zis, zqs);
}